// DCT2DLayer_1348619731263
// MI455X (gfx1250) — compile-verified
//
#include <hip/hip_runtime.h>

typedef __attribute__((ext_vector_type(2))) float v2f;
typedef __attribute__((ext_vector_type(8))) float v8f;

#define TILE_K 32
#define BLK_M  128
#define BLK_N  64
#define AP     34    // LDS pitch (floats) for A tile rows
#define BP     162   // LDS pitch (floats) for B k-pair rows (frag-native layout)
#define GK     512   // contraction length (both stages)
#define NT_K   (GK / TILE_K)

// Build the 512x512 DCT-II matrix: D[k][n] = cos(pi*(2n+1)k/1024)*sqrt(2/512),
// row 0 scaled by 1/sqrt(2). Exact integer mod-2048 range reduction keeps cosf
// accurate for all arguments.
__global__ __launch_bounds__(256) void dct_make(float* __restrict__ D) {
  int idx = blockIdx.x * 256 + threadIdx.x;
  int k = idx >> 9;
  int n = idx & 511;
  int r = ((2 * n + 1) * k) & 2047;               // (2n+1)k mod 2048
  float th = 3.14159265358979323846f * (float)r * (1.0f / 1024.0f);
  float v = cosf(th) * 0.0625f;                    // sqrt(2/512) = 1/16
  if (k == 0) v *= 0.70710678118654752f;
  D[idx] = v;
}

// ---- tile movement helpers -------------------------------------------------

__device__ __forceinline__ void fetchA(const float* __restrict__ Ab, int m0, int k0,
                                       long rsA, long csA, int tid, float (&r)[16]) {
#pragma unroll
  for (int t = 0; t < 16; ++t) {
    int idx = tid + t * 256;
    int kk = idx & (TILE_K - 1);
    int mm = idx >> 5;
    r[t] = Ab[(long)(m0 + mm) * rsA + (long)(k0 + kk) * csA];
  }
}

__device__ __forceinline__ void storeA(float* __restrict__ As, int tid,
                                       const float (&r)[16]) {
#pragma unroll
  for (int t = 0; t < 16; ++t) {
    int idx = tid + t * 256;
    int kk = idx & (TILE_K - 1);
    int mm = idx >> 5;
    As[mm * AP + kk] = r[t];
  }
}

__device__ __forceinline__ void fetchB(const float* __restrict__ Bb, int n0, int k0,
                                       long rsB, long csB, int tid, int bNFast,
                                       float (&r)[8]) {
  if (bNFast) {
#pragma unroll
    for (int t = 0; t < 8; ++t) {
      int idx = tid + t * 256;
      int nn = idx & (BLK_N - 1);
      int kk = idx >> 6;
      r[t] = Bb[(long)(k0 + kk) * rsB + (long)(n0 + nn) * csB];
    }
  } else {
#pragma unroll
    for (int t = 0; t < 8; ++t) {
      int idx = tid + t * 256;
      int kk = idx & (TILE_K - 1);
      int nn = idx >> 5;
      r[t] = Bb[(long)(k0 + kk) * rsB + (long)(n0 + nn) * csB];
    }
  }
}

// Fragment-native B layout: element (k, n) at (k/2)*BP + n*2 + (k&1), so a
// WMMA B fragment (K pair {2t,2t+1}, col n) is one aligned b64 in LDS.
__device__ __forceinline__ void storeB(float* __restrict__ Bs, int tid, int bNFast,
                                       const float (&r)[8]) {
  if (bNFast) {
#pragma unroll
    for (int t = 0; t < 8; ++t) {
      int idx = tid + t * 256;
      int nn = idx & (BLK_N - 1);
      int kk = idx >> 6;
      Bs[(kk >> 1) * BP + nn * 2 + (kk & 1)] = r[t];
    }
  } else {
#pragma unroll
    for (int t = 0; t < 8; ++t) {
      int idx = tid + t * 256;
      int kk = idx & (TILE_K - 1);
      int nn = idx >> 5;
      Bs[(kk >> 1) * BP + nn * 2 + (kk & 1)] = r[t];
    }
  }
}

// ---- one K-tile of WMMA math: 8 steps of K=4, 2x2 16x16 tiles per wave -----

__device__ __forceinline__ void mma_tile(const float* __restrict__ As,
                                         const float* __restrict__ Bs,
                                         v8f (&acc)[2][2],
                                         int wm, int wn, int row, int grp) {
#pragma unroll
  for (int kk = 0; kk < TILE_K; kk += 4) {
    v2f a[2], b[2];
#pragma unroll
    for (int mt = 0; mt < 2; ++mt)
      a[mt] = *(const v2f*)&As[(wm + mt * 16 + row) * AP + kk + 2 * grp];
#pragma unroll
    for (int nt = 0; nt < 2; ++nt)
      b[nt] = *(const v2f*)&Bs[((kk >> 1) + grp) * BP + (wn + nt * 16 + row) * 2];
#pragma unroll
    for (int mt = 0; mt < 2; ++mt)
#pragma unroll
      for (int nt = 0; nt < 2; ++nt)
        acc[mt][nt] = __builtin_amdgcn_wmma_f32_16x16x4_f32(
            false, a[mt], false, b[nt], (short)0, acc[mt][nt], false, false);
  }
}

// ---- generic strided GEMM, M = K = 512, double-buffered LDS pipeline -------

__global__ __launch_bounds__(256)
void dct_gemm(const float* __restrict__ A, const float* __restrict__ B,
              float* __restrict__ C, int N,
              long rsA, long csA, long rsB, long csB, long rsC, long csC,
              int zDiv, long zPitch, int useOffA, int useOffB, int useOffC,
              int bNFast) {
  (void)N;
  __shared__ float As[2][BLK_M * AP];
  __shared__ float Bs[2][(TILE_K / 2) * BP];

  const int tid = threadIdx.x;
  const int z   = blockIdx.z;
  const long off = (long)(z / zDiv) * zPitch + (long)(z % zDiv);
  const float* Ab = A + (useOffA ? off : 0);
  const float* Bb = B + (useOffB ? off : 0);
  float*       Cb = C + (useOffC ? off : 0);

  const int m0 = blockIdx.y * BLK_M;
  const int n0 = blockIdx.x * BLK_N;

  const int lane = tid & 31;
  const int wave = tid >> 5;
  const int wm   = (wave & 3) * 32;   // 4 waves along M
  const int wn   = (wave >> 2) * 32;  // 2 waves along N
  const int row  = lane & 15;
  const int grp  = lane >> 4;         // lane half: K pair select / M half

  v8f acc[2][2];
#pragma unroll
  for (int i = 0; i < 2; ++i)
#pragma unroll
    for (int j = 0; j < 2; ++j) acc[i][j] = (v8f)0.0f;

  float ra[16], rb[8];

  // prologue: tile 0 -> buffer 0
  fetchA(Ab, m0, 0, rsA, csA, tid, ra);
  fetchB(Bb, n0, 0, rsB, csB, tid, bNFast, rb);
  storeA(As[0], tid, ra);
  storeB(Bs[0], tid, bNFast, rb);
  __syncthreads();

  for (int t = 0; t < NT_K; ++t) {
    const int cur = t & 1;
    if (t + 1 < NT_K) {   // prefetch next tile into registers (overlaps WMMA)
      fetchA(Ab, m0, (t + 1) * TILE_K, rsA, csA, tid, ra);
      fetchB(Bb, n0, (t + 1) * TILE_K, rsB, csB, tid, bNFast, rb);
    }
    mma_tile(As[cur], Bs[cur], acc, wm, wn, row, grp);
    if (t + 1 < NT_K) {   // spill prefetch into the idle buffer
      storeA(As[cur ^ 1], tid, ra);
      storeB(Bs[cur ^ 1], tid, bNFast, rb);
    }
    __syncthreads();
  }

  // ---- store C fragments: lane holds N = lane&15, M = v + 8*grp
#pragma unroll
  for (int mt = 0; mt < 2; ++mt)
#pragma unroll
    for (int nt = 0; nt < 2; ++nt)
#pragma unroll
      for (int v = 0; v < 8; ++v) {
        int m = m0 + wm + mt * 16 + grp * 8 + v;
        int n = n0 + wn + nt * 16 + row;
        Cb[(long)m * rsC + (long)n * csC] = acc[mt][nt][v];
      }
}

extern "C" void kernel_launch(void* const* d_in, const int* in_sizes, int n_in,
                              void* d_out, int out_size, void* d_ws, size_t ws_size,
                              hipStream_t stream) {
  (void)in_sizes; (void)n_in; (void)out_size; (void)ws_size;
  const float* x = (const float*)d_in[0];
  float* out = (float*)d_out;
  float* Dm  = (float*)d_ws;          // 512*512 fp32 DCT matrix (1 MB)
  float* Y1  = Dm + 512 * 512;        // intermediate (32,512,512,3) fp32 (~100.7 MB)
  const long HWC = 512L * 512L * 3L;

  dct_make<<<dim3(1024), dim3(256), 0, stream>>>(Dm);

  // Stage 1: per batch b (z=0..31):  Y1_b(512 x 1536) = Dh(512x512) @ X_b(512x1536)
  dct_gemm<<<dim3(24, 4, 32), dim3(256), 0, stream>>>(
      Dm, x, Y1, 1536,
      /*rsA,csA*/ 512, 1, /*rsB,csB*/ 1536, 1, /*rsC,csC*/ 1536, 1,
      /*zDiv*/ 1, /*zPitch*/ HWC, /*useOffA/B/C*/ 0, 1, 1, /*bNFast*/ 1);

  // Stage 2: per (b,c) (z=0..95):  Out_bc(k x l) = Y1_bc(k x w) @ Dw^T(w x l)
  dct_gemm<<<dim3(8, 4, 96), dim3(256), 0, stream>>>(
      Y1, Dm, out, 512,
      /*rsA,csA*/ 1536, 3, /*rsB,csB*/ 1, 512, /*rsC,csC*/ 1536, 3,
      /*zDiv*/ 3, /*zPitch*/ HWC, /*useOffA/B/C*/ 1, 0, 1, /*bNFast*/ 0);
}